// GaussianRasterizer_72567767433661
// MI455X (gfx1250) — compile-verified
//
#include <hip/hip_runtime.h>

#define NG   512
#define WW   256
#define HH   256
#define FXc  256.0f
#define FYc  256.0f

typedef __attribute__((ext_vector_type(2))) float v2f;
typedef __attribute__((ext_vector_type(8))) float v8f;

// workspace layout (float offsets)
#define O_TZ   0
#define O_A    512
#define O_B    1024
#define O_C    1536
#define O_MX   2048
#define O_MY   2560
#define O_OP   3072
#define O_CR   3584
#define O_CG   4096
#define O_CB   4608
#define O_GF   5120      // 512 * 8 gaussian features (sorted)
#define O_SOP  9216
#define O_SCR  9728
#define O_SCG  10240
#define O_SCB  10752
#define O_SZ   11264

// ---------------------------------------------------------------------------
// Kernel 1: per-gaussian preprocessing
// ---------------------------------------------------------------------------
__global__ void gs_preprocess(const float* __restrict__ means3D,
                              const float* __restrict__ opac,
                              const float* __restrict__ cols,
                              const float* __restrict__ scales,
                              const float* __restrict__ rots,
                              const float* __restrict__ vm,   // 4x4 row-major
                              const float* __restrict__ pm,   // 4x4 row-major (V@P)
                              float* __restrict__ ws,
                              float* __restrict__ out_radii) {
    int i = blockIdx.x * blockDim.x + threadIdx.x;
    if (i >= NG) return;

    // quaternion -> rotation
    float qr = rots[i*4+0], qx = rots[i*4+1], qy = rots[i*4+2], qz = rots[i*4+3];
    float qn = rsqrtf(qr*qr + qx*qx + qy*qy + qz*qz);
    qr *= qn; qx *= qn; qy *= qn; qz *= qn;
    float R[3][3];
    R[0][0] = 1.f - 2.f*(qy*qy + qz*qz); R[0][1] = 2.f*(qx*qy - qr*qz); R[0][2] = 2.f*(qx*qz + qr*qy);
    R[1][0] = 2.f*(qx*qy + qr*qz); R[1][1] = 1.f - 2.f*(qx*qx + qz*qz); R[1][2] = 2.f*(qy*qz - qr*qx);
    R[2][0] = 2.f*(qx*qz - qr*qy); R[2][1] = 2.f*(qy*qz + qr*qx); R[2][2] = 1.f - 2.f*(qx*qx + qy*qy);

    float s[3] = { scales[i*3+0], scales[i*3+1], scales[i*3+2] };
    float M[3][3];
    #pragma unroll
    for (int r = 0; r < 3; ++r)
        #pragma unroll
        for (int c = 0; c < 3; ++c) M[r][c] = R[r][c] * s[c];
    float Sig[3][3];
    #pragma unroll
    for (int r = 0; r < 3; ++r)
        #pragma unroll
        for (int c = 0; c < 3; ++c)
            Sig[r][c] = M[r][0]*M[c][0] + M[r][1]*M[c][1] + M[r][2]*M[c][2];

    float mx3 = means3D[i*3+0], my3 = means3D[i*3+1], mz3 = means3D[i*3+2];
    // t = [x y z 1] @ V   (row-vector convention)
    float t[4];
    #pragma unroll
    for (int j = 0; j < 4; ++j)
        t[j] = mx3*vm[0*4+j] + my3*vm[1*4+j] + mz3*vm[2*4+j] + vm[3*4+j];
    float tz  = t[2];
    float tzc = fmaxf(tz, 1e-4f);
    const float limx = 1.3f * 0.5f, limy = 1.3f * 0.5f;
    float txc = fminf(fmaxf(t[0]/tzc, -limx), limx) * tzc;
    float tyc = fminf(fmaxf(t[1]/tzc, -limy), limy) * tzc;

    float inv_tz  = 1.f / tzc;
    float inv_tz2 = inv_tz * inv_tz;
    float J[2][3] = { { FXc*inv_tz, 0.f, -FXc*txc*inv_tz2 },
                      { 0.f, FYc*inv_tz, -FYc*tyc*inv_tz2 } };
    // T2 = J @ viewmatrix[:3,:3]^T  -> T2[r][k] = sum_j J[r][j] * vm[k*4+j]
    float T2[2][3];
    #pragma unroll
    for (int r = 0; r < 2; ++r)
        #pragma unroll
        for (int k = 0; k < 3; ++k)
            T2[r][k] = J[r][0]*vm[k*4+0] + J[r][1]*vm[k*4+1] + J[r][2]*vm[k*4+2];

    float tmp[2][3];
    #pragma unroll
    for (int r = 0; r < 2; ++r)
        #pragma unroll
        for (int k = 0; k < 3; ++k)
            tmp[r][k] = T2[r][0]*Sig[0][k] + T2[r][1]*Sig[1][k] + T2[r][2]*Sig[2][k];
    float c00 = tmp[0][0]*T2[0][0] + tmp[0][1]*T2[0][1] + tmp[0][2]*T2[0][2] + 0.3f;
    float c01 = tmp[0][0]*T2[1][0] + tmp[0][1]*T2[1][1] + tmp[0][2]*T2[1][2];
    float c11 = tmp[1][0]*T2[1][0] + tmp[1][1]*T2[1][1] + tmp[1][2]*T2[1][2] + 0.3f;

    float det = c00*c11 - c01*c01;
    float det_inv = 1.f / ((det != 0.f) ? det : 1.f);
    float A =  c11 * det_inv;
    float B = -c01 * det_inv;
    float C =  c00 * det_inv;
    float mid = 0.5f * (c00 + c11);
    float lam = mid + sqrtf(fmaxf(mid*mid - det, 0.1f));
    bool vis = (det > 0.f) && (tz > 0.2f);
    float rad = vis ? ceilf(3.f * sqrtf(lam)) : 0.f;

    // projection
    float h[4];
    #pragma unroll
    for (int j = 0; j < 4; ++j)
        h[j] = mx3*pm[0*4+j] + my3*pm[1*4+j] + mz3*pm[2*4+j] + pm[3*4+j];
    float pwv = 1.f / (h[3] + 1e-7f);
    float mx = ((h[0]*pwv + 1.f) * (float)WW - 1.f) * 0.5f;
    float my = ((h[1]*pwv + 1.f) * (float)HH - 1.f) * 0.5f;

    ws[O_TZ + i] = tz;
    ws[O_A  + i] = A;
    ws[O_B  + i] = B;
    ws[O_C  + i] = C;
    ws[O_MX + i] = mx;
    ws[O_MY + i] = my;
    ws[O_OP + i] = vis ? opac[i] : 0.f;
    ws[O_CR + i] = cols[i*3+0];
    ws[O_CG + i] = cols[i*3+1];
    ws[O_CB + i] = cols[i*3+2];
    out_radii[i] = rad;
}

// ---------------------------------------------------------------------------
// Kernel 2: stable rank-sort by depth + scatter of WMMA gaussian features
// ---------------------------------------------------------------------------
__global__ void gs_sort(float* __restrict__ ws) {
    __shared__ float stz[NG];
    int i = threadIdx.x;
    stz[i] = ws[O_TZ + i];
    __syncthreads();
    float ti = stz[i];
    int rank = 0;
    for (int j = 0; j < NG; ++j) {
        float tj = stz[j];
        rank += (tj < ti) || (tj == ti && j < i);
    }
    float A  = ws[O_A + i],  B = ws[O_B + i],  C = ws[O_C + i];
    float mx = ws[O_MX + i], my = ws[O_MY + i];
    float* gf = ws + O_GF + rank*8;
    gf[0] = -0.5f*(A*mx*mx + C*my*my) - B*mx*my;  // const
    gf[1] =  A*mx + B*my;                          // * px
    gf[2] = -0.5f*A;                               // * px^2
    gf[3] =  C*my + B*mx;                          // * py
    gf[4] = -0.5f*C;                               // * py^2
    gf[5] = -B;                                    // * px*py
    gf[6] = 0.f;
    gf[7] = 0.f;
    ws[O_SOP + rank] = ws[O_OP + i];
    ws[O_SCR + rank] = ws[O_CR + i];
    ws[O_SCG + rank] = ws[O_CG + i];
    ws[O_SCB + rank] = ws[O_CB + i];
    ws[O_SZ  + rank] = ti;
}

// ---------------------------------------------------------------------------
// Kernel 3: WMMA rasterizer. Tile: M = 16 gaussians, N = 16 pixels, K = 8.
// power(16g x 16p) via two V_WMMA_F32_16X16X4_F32; per-lane serial blend of
// 8 depth-ordered gaussians; lane halves merged with shfl_xor(16).
// ---------------------------------------------------------------------------
__global__ __launch_bounds__(128) void gs_raster(const float* __restrict__ ws,
                                                 const float* __restrict__ bg,
                                                 float* __restrict__ out) {
    __shared__ float sGf[NG*8];
    __shared__ float sOp[NG], sCr[NG], sCg[NG], sCb[NG], sZ[NG];
    int tid = threadIdx.x;
    for (int i = tid; i < NG*8; i += 128) sGf[i] = ws[O_GF + i];
    for (int i = tid; i < NG;   i += 128) {
        sOp[i] = ws[O_SOP + i];
        sCr[i] = ws[O_SCR + i];
        sCg[i] = ws[O_SCG + i];
        sCb[i] = ws[O_SCB + i];
        sZ [i] = ws[O_SZ  + i];
    }
    __syncthreads();

    int lane = tid & 31;
    int wave = tid >> 5;
    int n    = lane & 15;       // pixel column of tile (N)
    int half = lane >> 4;       // 0: K=0..1 / M rows 0..7,  1: K=2..3 / M rows 8..15
    int p    = (blockIdx.x * 4 + wave) * 16 + n;   // linear pixel id
    float fx = (float)(p & (WW - 1));
    float fy = (float)(p >> 8);
    float f[8] = { 1.f, fx, fx*fx, fy, fy*fy, fx*fy, 0.f, 0.f };
    int k0 = half * 2;
    v2f b1; b1.x = f[k0];     b1.y = f[k0+1];
    v2f b2; b2.x = f[k0+4];   b2.y = f[k0+5];

    float runT = 1.f, rc0 = 0.f, rc1 = 0.f, rc2 = 0.f, rd = 0.f;

    for (int t = 0; t < NG/16; ++t) {
        // A-matrix: lane holds gaussian M = lane&15, K pair selected by half
        const float* g8 = &sGf[(t*16 + n) * 8];
        v2f a1; a1.x = g8[k0];   a1.y = g8[k0+1];
        v2f a2; a2.x = g8[k0+4]; a2.y = g8[k0+5];
        v8f c = {0.f,0.f,0.f,0.f,0.f,0.f,0.f,0.f};
        c = __builtin_amdgcn_wmma_f32_16x16x4_f32(false, a1, false, b1,
                                                  (short)0, c, false, false);
        c = __builtin_amdgcn_wmma_f32_16x16x4_f32(false, a2, false, b2,
                                                  (short)0, c, false, false);
        // per-lane serial blend of 8 depth-ordered gaussians
        int gb = t*16 + half*8;
        float lT = 1.f, l0 = 0.f, l1 = 0.f, l2 = 0.f, ld = 0.f;
        #pragma unroll
        for (int r = 0; r < 8; ++r) {
            float pw = c[r];
            int   g  = gb + r;
            float al = fminf(0.99f, sOp[g] * __expf(pw));
            bool  ok = (pw <= 0.f) && (al >= 0.0039215686f);
            al = ok ? al : 0.f;
            float w = al * lT;
            l0 += w * sCr[g];
            l1 += w * sCg[g];
            l2 += w * sCb[g];
            ld += w * sZ[g];
            lT *= (1.f - al);
        }
        // merge lane halves: segmented transmittance composition
        float oT = __shfl_xor(lT, 16, 32);
        float o0 = __shfl_xor(l0, 16, 32);
        float o1 = __shfl_xor(l1, 16, 32);
        float o2 = __shfl_xor(l2, 16, 32);
        float od = __shfl_xor(ld, 16, 32);
        float loT = half ? oT : lT, hiT = half ? lT : oT;
        float lo0 = half ? o0 : l0, hi0 = half ? l0 : o0;
        float lo1 = half ? o1 : l1, hi1 = half ? l1 : o1;
        float lo2 = half ? o2 : l2, hi2 = half ? l2 : o2;
        float lod = half ? od : ld, hid = half ? ld : od;
        float tc0 = lo0 + loT * hi0;
        float tc1 = lo1 + loT * hi1;
        float tc2 = lo2 + loT * hi2;
        float tcd = lod + loT * hid;
        float tT  = loT * hiT;
        rc0 += runT * tc0;
        rc1 += runT * tc1;
        rc2 += runT * tc2;
        rd  += runT * tcd;
        runT *= tT;
    }

    rc0 += runT * bg[0];
    rc1 += runT * bg[1];
    rc2 += runT * bg[2];

    if (half == 0) {
        out[0*65536 + p] = rc0;   // color R plane
        out[1*65536 + p] = rc1;   // color G plane
        out[2*65536 + p] = rc2;   // color B plane
        out[3*65536 + p] = rd;    // depth
        out[4*65536 + p] = runT;  // Tfinal
    }
}

// ---------------------------------------------------------------------------
extern "C" void kernel_launch(void* const* d_in, const int* in_sizes, int n_in,
                              void* d_out, int out_size, void* d_ws, size_t ws_size,
                              hipStream_t stream) {
    (void)in_sizes; (void)n_in; (void)out_size; (void)ws_size;
    const float* means3D = (const float*)d_in[0];
    const float* opac    = (const float*)d_in[1];
    const float* cols    = (const float*)d_in[2];
    const float* scales  = (const float*)d_in[3];
    const float* rots    = (const float*)d_in[4];
    const float* bg      = (const float*)d_in[5];
    const float* vm      = (const float*)d_in[6];
    const float* pm      = (const float*)d_in[7];
    float* out = (float*)d_out;
    float* ws  = (float*)d_ws;

    gs_preprocess<<<2, 256, 0, stream>>>(means3D, opac, cols, scales, rots,
                                         vm, pm, ws, out + 5*65536);
    gs_sort<<<1, NG, 0, stream>>>(ws);
    gs_raster<<<1024, 128, 0, stream>>>(ws, bg, out);
}